// SelectiveSSM_33904471834645
// MI455X (gfx1250) — compile-verified
//
#include <hip/hip_runtime.h>
#include <hip/hip_bf16.h>
#include <math.h>

typedef __attribute__((ext_vector_type(2))) float v2f;
typedef __attribute__((ext_vector_type(8))) float v8f;

#define D_IN   1024
#define D_ST   16
#define DT_RANK 64
#define N_E    96            // DT_RANK + 2*D_STATE
#define BATCH  2
#define SEQ    2048
#define M_TOT  (BATCH*SEQ)   // 4096
#define NCHUNK 8
#define LCHUNK (SEQ/NCHUNK)  // 256
#define NCHAN  (BATCH*D_IN)  // 2048

// ---------------------------------------------------------------------------
// Kernel 1: x_dbl[4096,96] = X[4096,1024] @ W_xproj[96,1024]^T  (fp32 WMMA)
// One wave per 16x16 output tile; K-loop of V_WMMA_F32_16X16X4_F32.
// A frag: lane holds A[m0+(lane&15)][k + (lane>>4)*2 + {0,1}]  (float2)
// B frag: lane holds W[n0+(lane&15)][k + (lane>>4)*2 + {0,1}]  (float2)
// D frag: vgpr v -> (m0 + v + 8*(lane>>4), n0 + (lane&15))
// ---------------------------------------------------------------------------
__global__ void xproj_wmma(const float* __restrict__ X,
                           const float* __restrict__ W,
                           float* __restrict__ xdbl) {
    int wave = (blockIdx.x * blockDim.x + threadIdx.x) >> 5;
    int lane = threadIdx.x & 31;
    const int NT = N_E / 16;                 // 6 N-tiles
    int nt = wave % NT;
    int mt = wave / NT;
    if (mt >= M_TOT / 16) return;            // wave-uniform guard
    int m0 = mt * 16, n0 = nt * 16;
    int half = lane >> 4, lq = lane & 15;
    int koff = half * 2;
    const float* aptr = X + (size_t)(m0 + lq) * D_IN + koff;
    const float* bptr = W + (size_t)(n0 + lq) * D_IN + koff;
    v8f acc = {};
    for (int k = 0; k < D_IN; k += 4) {
        v2f a = *(const v2f*)(aptr + k);
        v2f b = *(const v2f*)(bptr + k);
        acc = __builtin_amdgcn_wmma_f32_16x16x4_f32(
            false, a, false, b, (short)0, acc, false, false);
    }
    int n = n0 + lq;
#pragma unroll
    for (int v = 0; v < 8; ++v) {
        int m = m0 + v + 8 * half;
        xdbl[(size_t)m * N_E + n] = acc[v];
    }
}

// ---------------------------------------------------------------------------
// Kernel 2: dt[4096,1024] = softplus(x_dbl[:, :64] @ W_dt[1024,64]^T + b_dt)
// ---------------------------------------------------------------------------
__global__ void dt_wmma(const float* __restrict__ xdbl,
                        const float* __restrict__ Wdt,
                        const float* __restrict__ bdt,
                        float* __restrict__ dt) {
    int wave = (blockIdx.x * blockDim.x + threadIdx.x) >> 5;
    int lane = threadIdx.x & 31;
    const int NT = D_IN / 16;                // 64 N-tiles
    int nt = wave % NT;
    int mt = wave / NT;
    if (mt >= M_TOT / 16) return;            // wave-uniform guard
    int m0 = mt * 16, n0 = nt * 16;
    int half = lane >> 4, lq = lane & 15;
    int koff = half * 2;
    const float* aptr = xdbl + (size_t)(m0 + lq) * N_E + koff;   // dt_low part
    const float* bptr = Wdt + (size_t)(n0 + lq) * DT_RANK + koff;
    v8f acc = {};
#pragma unroll
    for (int k = 0; k < DT_RANK; k += 4) {
        v2f a = *(const v2f*)(aptr + k);
        v2f b = *(const v2f*)(bptr + k);
        acc = __builtin_amdgcn_wmma_f32_16x16x4_f32(
            false, a, false, b, (short)0, acc, false, false);
    }
    int n = n0 + lq;
    float bias = bdt[n];
#pragma unroll
    for (int v = 0; v < 8; ++v) {
        int m = m0 + v + 8 * half;
        float z = acc[v] + bias;
        float sp = (z > 20.0f) ? z : log1pf(__expf(z));   // stable softplus
        dt[(size_t)m * D_IN + n] = sp;
    }
}

// ---------------------------------------------------------------------------
// Kernel 3 (phase 1): per (channel, chunk, state) local scan -> (prod A, h_end)
// 16 lanes (= states) per unit; 2 units per wave32.
// ---------------------------------------------------------------------------
__global__ void scan_phase1(const float* __restrict__ X,
                            const float* __restrict__ xdbl,
                            const float* __restrict__ dt,
                            const float* __restrict__ A_log,
                            float* __restrict__ aprod,
                            float* __restrict__ hend) {
    int gid = blockIdx.x * blockDim.x + threadIdx.x;
    int unit = gid >> 4;
    int s = gid & 15;
    if (unit >= NCHAN * NCHUNK) return;
    int chunk = unit % NCHUNK;
    int channel = unit / NCHUNK;
    int b = channel / D_IN;
    int d = channel % D_IN;
    float A = -__expf(A_log[d * D_ST + s]);
    float ap = 1.0f, h = 0.0f;
    int l0 = chunk * LCHUNK;
    for (int l = l0; l < l0 + LCHUNK; ++l) {
        int m = b * SEQ + l;
        float dtv = dt[(size_t)m * D_IN + d];
        float xv  = X[(size_t)m * D_IN + d];
        float Bm  = xdbl[(size_t)m * N_E + DT_RANK + s];
        float ab  = __expf(dtv * A);
        h  = ab * h + dtv * Bm * xv;
        ap *= ab;
    }
    int idx = (channel * D_ST + s) * NCHUNK + chunk;
    aprod[idx] = ap;
    hend[idx]  = h;
}

// ---------------------------------------------------------------------------
// Kernel 4 (phase 2): length-8 exclusive cross-chunk combine per (channel,s)
// ---------------------------------------------------------------------------
__global__ void scan_phase2(const float* __restrict__ aprod,
                            const float* __restrict__ hend,
                            float* __restrict__ carry) {
    int t = blockIdx.x * blockDim.x + threadIdx.x;   // channel*16 + s
    if (t >= NCHAN * D_ST) return;
    float c = 0.0f;
#pragma unroll
    for (int ch = 0; ch < NCHUNK; ++ch) {
        int idx = t * NCHUNK + ch;
        carry[idx] = c;
        c = aprod[idx] * c + hend[idx];
    }
}

// ---------------------------------------------------------------------------
// Kernel 5 (phase 3): replay local scan with carry-in, reduce 16 states via
// __shfl_xor within each 16-lane half-wave, write y = sum_s h*C + D*x.
// ---------------------------------------------------------------------------
__global__ void scan_phase3(const float* __restrict__ X,
                            const float* __restrict__ xdbl,
                            const float* __restrict__ dt,
                            const float* __restrict__ A_log,
                            const float* __restrict__ Dp,
                            const float* __restrict__ carry,
                            float* __restrict__ y) {
    int gid = blockIdx.x * blockDim.x + threadIdx.x;
    int unit = gid >> 4;
    int s = gid & 15;
    if (unit >= NCHAN * NCHUNK) return;
    int chunk = unit % NCHUNK;
    int channel = unit / NCHUNK;
    int b = channel / D_IN;
    int d = channel % D_IN;
    float A = -__expf(A_log[d * D_ST + s]);
    float h = carry[(channel * D_ST + s) * NCHUNK + chunk];
    float Dd = Dp[d];
    int l0 = chunk * LCHUNK;
    for (int l = l0; l < l0 + LCHUNK; ++l) {
        int m = b * SEQ + l;
        float dtv = dt[(size_t)m * D_IN + d];
        float xv  = X[(size_t)m * D_IN + d];
        float Bm  = xdbl[(size_t)m * N_E + DT_RANK + s];
        float Cm  = xdbl[(size_t)m * N_E + DT_RANK + D_ST + s];
        float ab  = __expf(dtv * A);
        h = ab * h + dtv * Bm * xv;
        float contrib = h * Cm;
#pragma unroll
        for (int msk = 1; msk < 16; msk <<= 1)       // stays inside 16-lane half
            contrib += __shfl_xor(contrib, msk, 32);
        if (s == 0)
            y[(size_t)m * D_IN + d] = contrib + Dd * xv;
    }
}

// ---------------------------------------------------------------------------
extern "C" void kernel_launch(void* const* d_in, const int* in_sizes, int n_in,
                              void* d_out, int out_size, void* d_ws, size_t ws_size,
                              hipStream_t stream) {
    const float* x       = (const float*)d_in[0];   // (2,2048,1024)
    const float* A_log   = (const float*)d_in[1];   // (1024,16)
    const float* Dp      = (const float*)d_in[2];   // (1024,)
    const float* W_xproj = (const float*)d_in[3];   // (96,1024)
    const float* W_dt    = (const float*)d_in[4];   // (1024,64)
    const float* b_dt    = (const float*)d_in[5];   // (1024,)
    float* y = (float*)d_out;

    char* ws = (char*)d_ws;
    size_t off = 0;
    float* xdbl  = (float*)(ws + off); off += (size_t)M_TOT * N_E * 4;          // 1.5 MB
    off = (off + 255) & ~(size_t)255;
    float* dtbuf = (float*)(ws + off); off += (size_t)M_TOT * D_IN * 4;         // 16 MB
    off = (off + 255) & ~(size_t)255;
    float* aprod = (float*)(ws + off); off += (size_t)NCHAN * D_ST * NCHUNK * 4;
    off = (off + 255) & ~(size_t)255;
    float* hend  = (float*)(ws + off); off += (size_t)NCHAN * D_ST * NCHUNK * 4;
    off = (off + 255) & ~(size_t)255;
    float* carry = (float*)(ws + off);

    // K1: (4096/16)*(96/16) = 1536 tiles, 4 waves/block
    xproj_wmma<<<1536 / 4, 128, 0, stream>>>(x, W_xproj, xdbl);
    // K2: (4096/16)*(1024/16) = 16384 tiles, 8 waves/block
    dt_wmma<<<16384 / 8, 256, 0, stream>>>(xdbl, W_dt, b_dt, dtbuf);
    // K3: 2048 channels * 8 chunks * 16 states = 262144 lanes
    scan_phase1<<<(NCHAN * NCHUNK * 16) / 256, 256, 0, stream>>>(
        x, xdbl, dtbuf, A_log, aprod, hend);
    // K4: 32768 (channel,state) threads
    scan_phase2<<<(NCHAN * D_ST) / 256, 256, 0, stream>>>(aprod, hend, carry);
    // K5: same shape as K3
    scan_phase3<<<(NCHAN * NCHUNK * 16) / 256, 256, 0, stream>>>(
        x, xdbl, dtbuf, A_log, Dp, carry, y);
}